// KNN_11003706212687
// MI455X (gfx1250) — compile-verified
//
#include <hip/hip_runtime.h>

typedef float v2f __attribute__((ext_vector_type(2)));
typedef float v4f __attribute__((ext_vector_type(4)));
typedef float v8f __attribute__((ext_vector_type(8)));

#define BATCH 4
#define NPTS 8192
#define NQRY 8192
#define KNN 16
#define WGSIZE 256
#define WAVES_PER_WG 8
#define QRY_PER_WG (WAVES_PER_WG * 16)      // 128 queries per workgroup
#define WGS_PER_BATCH (NQRY / QRY_PER_WG)   // 64

__global__ __launch_bounds__(WGSIZE) void knn_wmma_kernel(
    const float* __restrict__ xyz, const float* __restrict__ new_xyz,
    int* __restrict__ out)
{
  // Transformed points for this batch: (-2x, -2y, -2z, x^2+y^2+z^2). 128 KB.
  __shared__ v4f pts[NPTS];

  const int b    = blockIdx.x / WGS_PER_BATCH;
  const int gq   = blockIdx.x % WGS_PER_BATCH;
  const int tid  = threadIdx.x;
  const int wave = tid >> 5;
  const int lane = tid & 31;
  const int half = lane >> 4;   // 0: lanes 0-15 (K=0,1 / points m+0..7), 1: lanes 16-31
  const int l16  = lane & 15;

  // --- Stage transformed points into LDS (whole batch, once per WG) ---
  for (int n = tid; n < NPTS; n += WGSIZE) {
    const float* p = xyz + ((size_t)b * NPTS + n) * 3;
    float x = p[0], y = p[1], z = p[2];
    v4f t;
    t.x = -2.0f * x;
    t.y = -2.0f * y;
    t.z = -2.0f * z;
    t.w = x * x + y * y + z * z;
    pts[n] = t;
  }
  __syncthreads();

  // --- Loop-invariant B operand: query q = (qx, qy, qz, 1) ---
  // B is 4x16 (KxN): lanes 0-15 hold K=0,1 ; lanes 16-31 hold K=2,3 ; N = lane%16
  const int q = gq * QRY_PER_WG + wave * 16 + l16;
  const float* qp = new_xyz + ((size_t)b * NQRY + q) * 3;
  v2f bq;
  if (half == 0) { bq.x = qp[0]; bq.y = qp[1]; }
  else           { bq.x = qp[2]; bq.y = 1.0f;  }

  // --- Running top-K (ascending; bd[15] = current worst kept) ---
  float bd[KNN];
  int   bi[KNN];
  #pragma unroll
  for (int t = 0; t < KNN; ++t) { bd[t] = __builtin_huge_valf(); bi[t] = 0; }

  const v2f* ap = reinterpret_cast<const v2f*>(pts);
  const int ebase = half * 8;   // this lane sees points n0 + ebase + 0..7

  // --- Scan all points in 16-wide tiles: one V_WMMA_F32_16X16X4_F32 per tile.
  // A-operand for the next tile is prefetched (branch-free wrap) so the
  // ds_load latency hides under the selection VALU work.
  v2f av = ap[l16 * 2 + half];
  for (int n0 = 0; n0 < NPTS; n0 += 16) {
    v8f z = {};
    v8f acc = __builtin_amdgcn_wmma_f32_16x16x4_f32(
        false, av, false, bq, (short)0, z, false, false);

    int nn = (n0 + 16) & (NPTS - 1);          // wraps to 0 on last tile (harmless)
    av = ap[(nn + l16) * 2 + half];

    #pragma unroll
    for (int e = 0; e < 8; ++e) {
      float v = acc[e];
      if (v < bd[KNN - 1]) {      // wave-level skip via v_cmpx + s_cbranch_execz
        // Branch-free bubble insertion: carry (cv,ci) down the sorted list.
        // All four updates select on the same v_cmp -> 1 cmp + 4 cndmask per
        // slot, no IEEE min/max canonicalization ops.
        float cv = v;
        int   ci = n0 + ebase + e;
        #pragma unroll
        for (int j = 0; j < KNN; ++j) {
          bool  c  = cv < bd[j];
          float lo = c ? cv    : bd[j];
          float hi = c ? bd[j] : cv;
          int   ni = c ? ci    : bi[j];
          int   xi = c ? bi[j] : ci;
          bd[j] = lo;
          bi[j] = ni;
          cv = hi;
          ci = xi;
        }
      }
    }
  }

  __syncthreads();   // all waves done reading pts -> safe to reuse LDS as merge scratch

  // Per-wave scratch inside the (now free) pts region: 1024 dwords per wave.
  float* scratch_d = reinterpret_cast<float*>(pts) + wave * 1024;
  int*   scratch_i = reinterpret_cast<int*>(scratch_d + 512);
  #pragma unroll
  for (int t = 0; t < KNN; ++t) {
    scratch_d[lane * KNN + t] = bd[t];
    scratch_i[lane * KNN + t] = bi[t];
  }
  __syncthreads();

  // --- Merge the two sorted half-lists (lane L covers n%16 in 0..7, L+16 in 8..15) ---
  if (half == 0) {
    const float* dlo = scratch_d + l16 * KNN;
    const float* dhi = scratch_d + (l16 + 16) * KNN;
    const int*   ilo = scratch_i + l16 * KNN;
    const int*   ihi = scratch_i + (l16 + 16) * KNN;
    int* op = out + ((size_t)b * NQRY + q) * KNN;
    int pi = 0, pj = 0;
    #pragma unroll
    for (int t = 0; t < KNN; ++t) {   // pi+pj == t <= 15, so reads stay in range
      float da = dlo[pi], db = dhi[pj];
      int   ia = ilo[pi], ib = ihi[pj];
      bool ta = (da < db) || ((da == db) && (ia < ib));  // stable tie-break: lower index
      op[t] = ta ? ia : ib;
      pi += ta ? 1 : 0;
      pj += ta ? 0 : 1;
    }
  }
}

extern "C" void kernel_launch(void* const* d_in, const int* in_sizes, int n_in,
                              void* d_out, int out_size, void* d_ws, size_t ws_size,
                              hipStream_t stream) {
  (void)in_sizes; (void)n_in; (void)d_ws; (void)ws_size; (void)out_size;
  const float* xyz     = (const float*)d_in[0];
  const float* new_xyz = (const float*)d_in[1];
  int* out = (int*)d_out;
  dim3 grid(BATCH * WGS_PER_BATCH);
  dim3 block(WGSIZE);
  hipLaunchKernelGGL(knn_wmma_kernel, grid, block, 0, stream, xyz, new_xyz, out);
}